// Spline_38843684225892
// MI455X (gfx1250) — compile-verified
//
#include <hip/hip_runtime.h>
#include <hip/hip_bf16.h>
#include <cstddef>

typedef float v2f __attribute__((ext_vector_type(2)));
typedef float v8f __attribute__((ext_vector_type(8)));

#define NI   256   // num intervals
#define NB   16    // batch
#define NDIM 3
#define NCOL (NB * NDIM)   // 48 output columns per t

// ---------------------------------------------------------------------------
// Kernel 1: coeff[i][col][p] = (BEZIER_M @ points)[b,i,p,d], col = b*3+d.
// 256*48 = 12288 threads, 192KB table in d_ws. L2-resident afterwards.
// ---------------------------------------------------------------------------
__global__ void spline_coeff_kernel(const float* __restrict__ joint,
                                    const float* __restrict__ ctrl,
                                    float* __restrict__ cw) {
    int idx = blockIdx.x * blockDim.x + threadIdx.x;
    if (idx >= NI * NCOL) return;
    int i   = idx / NCOL;
    int col = idx - i * NCOL;
    int b   = col / NDIM;
    int d   = col - b * NDIM;

    float P0 = joint[((size_t)b * (NI + 1) + i)     * NDIM + d];
    float P1 = ctrl [((size_t)b * (NI * 2) + 2 * i) * NDIM + d];
    float P2 = ctrl [((size_t)b * (NI * 2) + 2 * i + 1) * NDIM + d];
    float P3 = joint[((size_t)b * (NI + 1) + i + 1) * NDIM + d];

    // BEZIER_M rows
    float c0 = P0;
    float c1 = -3.0f * P0 + 3.0f * P1;
    float c2 =  3.0f * P0 - 6.0f * P1 + 3.0f * P2;
    float c3 = -P0 + 3.0f * P1 - 3.0f * P2 + P3;

    float* o = cw + (size_t)idx * 4;
    o[0] = c0; o[1] = c1; o[2] = c2; o[3] = c3;
}

__device__ __forceinline__ int interval_of(float tv) {
    tv = fminf(tv, 1.0f - 1e-10f);
    int iv = (int)floorf((float)NI * tv);
    iv = iv < 0 ? 0 : (iv > NI - 1 ? NI - 1 : iv);
    return iv;
}

// ---------------------------------------------------------------------------
// Kernel 2: one wave32 handles a tile of 16 consecutive t values.
// Uniform-interval tile (common case, t sorted): 3x V_WMMA_F32_16X16X4_F32
//   A (16x4)  = [1, pt, pt^2, pt^3] per t-row
//   B (4x48)  = coeff slab of the shared interval (16 cols per WMMA)
// Mixed tile / ragged tail: scalar 4-FMA fallback per output.
// ---------------------------------------------------------------------------
__global__ void spline_eval_kernel(const float* __restrict__ t,
                                   const float* __restrict__ cw,
                                   float* __restrict__ out,
                                   int NT) {
    const int lane = threadIdx.x & 31;
    const int wave = blockIdx.x * (blockDim.x >> 5) + (threadIdx.x >> 5);
    const int base = wave * 16;
    if (base >= NT) return;

    const int  m  = lane & 15;
    const bool hi = lane >= 16;

    const bool full = (base + 16 <= NT);
    const int  ti_own = base + (full ? m : min(m, NT - 1 - base));

    // per-lane t -> interval + local poly parameter
    float tm = fminf(t[ti_own], 1.0f - 1e-10f);
    float u  = (float)NI * tm;
    int   iv = interval_of(tm);
    float pt = u - (float)iv;

    // wave-uniform tile classification from sorted endpoints (scalar loads)
    bool uniform = false;
    if (full) {
        int i0  = interval_of(t[base]);
        int i15 = interval_of(t[base + 15]);
        uniform = (i0 == i15);
    }

    if (uniform) {
        // ---- WMMA fast path (EXEC all-ones: branch is wave-uniform) ----
        const int i = iv;   // same for all lanes in tile
        float pt2 = pt * pt;
        v2f a;
        a.x = hi ? pt2       : 1.0f;   // K = 0 / 2
        a.y = hi ? pt2 * pt  : pt;     // K = 1 / 3

        const int tro = hi ? 8 : 0;    // D-matrix row offset for high half
#pragma unroll
        for (int g = 0; g < 3; ++g) {
            const int col = g * 16 + m;
            // B fragment: rows (K) 0,1 for low lanes, 2,3 for high lanes
            const v2f bv =
                *(const v2f*)(cw + ((size_t)i * NCOL + col) * 4 + (hi ? 2 : 0));
            v8f c = {};
            c = __builtin_amdgcn_wmma_f32_16x16x4_f32(
                    /*neg_a=*/false, a, /*neg_b=*/false, bv,
                    /*c_mod=*/(short)0, c, /*reuse_a=*/false, /*reuse_b=*/false);

            const int bb = col / NDIM;
            const int dd = col - bb * NDIM;
            float* op = out + (size_t)bb * NT * NDIM + dd;
#pragma unroll
            for (int r = 0; r < 8; ++r) {
                const int ti = base + tro + r;    // D: VGPR r -> rows r / r+8
                op[(size_t)ti * NDIM] = c[r];
            }
        }
    } else {
        // ---- scalar fallback: lane pair (l, l+16) splits the 48 columns ----
        if (base + m >= NT) return;
        const float p1 = pt, p2 = pt * pt, p3 = p2 * pt;
        const float* cb = cw + (size_t)iv * NCOL * 4;
        const int cbase = hi ? 24 : 0;
        const size_t trow = (size_t)(base + m) * NDIM;
#pragma unroll 4
        for (int j = 0; j < 24; ++j) {
            const int col = cbase + j;
            const float* cc = cb + (size_t)col * 4;
            float v = cc[0] + cc[1] * p1 + cc[2] * p2 + cc[3] * p3;
            const int bb = col / NDIM;
            const int dd = col - bb * NDIM;
            out[(size_t)bb * NT * NDIM + trow + dd] = v;
        }
    }
}

extern "C" void kernel_launch(void* const* d_in, const int* in_sizes, int n_in,
                              void* d_out, int out_size, void* d_ws, size_t ws_size,
                              hipStream_t stream) {
    const float* t     = (const float*)d_in[0];
    const float* joint = (const float*)d_in[1];
    const float* ctrl  = (const float*)d_in[2];
    float* out = (float*)d_out;
    float* cw  = (float*)d_ws;          // 256*48*4 floats = 192 KB
    const int NT = in_sizes[0];

    // 1) build coefficient table (tiny)
    {
        const int n = NI * NCOL;
        spline_coeff_kernel<<<(n + 255) / 256, 256, 0, stream>>>(joint, ctrl, cw);
    }
    // 2) evaluate: one wave32 per 16 t-values
    {
        const int waves   = (NT + 15) / 16;
        const int threads = waves * 32;
        spline_eval_kernel<<<(threads + 255) / 256, 256, 0, stream>>>(t, cw, out, NT);
    }
}